// LowPassFilter_30580167147905
// MI455X (gfx1250) — compile-verified
//
#include <hip/hip_runtime.h>

// ---------------------------------------------------------------------------
// First-order IIR low-pass, y_t = a*x_t + (1-a)*y_{t-1}, over x[16,65536,64].
//
// Memory-bound (512MB traffic ~22us @ 23.3TB/s; +25% warm-up reads ~25us).
//  * Chunk T into 512-step tiles per wave, 128-step warm-up from zero state
//    (b^128 ~ 4e-19 << fp32 ulp of output) -> 8192 independent waves.
//  * 16 timesteps x 16 channels per step as a lower-triangular 16x16 matmul:
//    4 chained V_WMMA_F32_16X16X4_F32 (K=16), rank-1 carry fix-up (8 FMAs),
//    state carried via ds_bpermute of C row 15 (VGPR7, lanes 16..31).
//  * Row addressing via running pointers + constant immediate offsets;
//    main loop software-pipelined (prefetch next block's B fragments).
//  * Wave-uniform ids forced scalar via readfirstlane -> SALU loop control,
//    EXEC stays all-ones around every WMMA.
// ---------------------------------------------------------------------------

typedef __attribute__((ext_vector_type(2))) float v2f;
typedef __attribute__((ext_vector_type(8))) float v8f;

namespace {
constexpr double kPI    = 3.14159265358979323846;
constexpr double kDT    = 1.0 / 16000.0;
constexpr double kTAU   = 1.0 / (2.0 * kPI * 1000.0);
constexpr float  kALPHA = (float)(kDT / (kDT + kTAU));   // ~0.28197

constexpr int kB     = 16;
constexpr int kT     = 65536;
constexpr int kC     = 64;
constexpr int kCHUNK = 512;           // output samples per wave
constexpr int kWARM  = 128;           // warm-up samples
constexpr int kNCH   = kT / kCHUNK;   // 128 chunks
constexpr int kCT    = kC / 16;       // 4 channel tiles
constexpr int kBLOCKS = (kB * kNCH * kCT * 32) / 256;   // 1024
}

// Load one 16x16 B tile's fragments for this lane.
// p already includes (c0 + col + 2*half*kC); row offsets are immediates.
__device__ __forceinline__ void loadB(v2f bf[4], const float* __restrict__ p) {
#pragma unroll
    for (int kk = 0; kk < 4; ++kk) {
        bf[kk].x = p[(size_t)(4 * kk) * kC];
        bf[kk].y = p[(size_t)(4 * kk + 1) * kC];
    }
}

__global__ __launch_bounds__(256) void
LowPassFilter_30580167147905_kernel(const float* __restrict__ x,
                                    float* __restrict__ y) {
    __shared__ float Tm[16 * 16];   // T[i][j] = a * b^(i-j), lower triangular

    const float a  = kALPHA;
    const float bc = 1.0f - kALPHA;

    const int tid = threadIdx.x;
    {
        const int i = tid >> 4, j = tid & 15;
        float v = 0.0f;
        if (i >= j) {
            float p = a;
            for (int k = 0; k < i - j; ++k) p *= bc;
            v = p;
        }
        Tm[i * 16 + j] = v;
    }
    __syncthreads();

    const int lane = tid & 31;
    const int half = lane >> 4;
    const int col  = lane & 15;

    // Wave-uniform ids, forced into SGPRs so loop control is scalar and EXEC
    // is never masked around the WMMAs.
    const int wave = __builtin_amdgcn_readfirstlane((blockIdx.x * blockDim.x + tid) >> 5);
    const int ct   = wave & (kCT - 1);
    const int q    = (wave >> 2) & (kNCH - 1);
    const int bidx = wave >> 9;

    // A fragments: 16x4 fp32 layout (lane M = lane&15, K = vgpr + 2*half).
    v2f afr[4];
#pragma unroll
    for (int kk = 0; kk < 4; ++kk) {
        const int k0 = 4 * kk + 2 * half;
        afr[kk].x = Tm[col * 16 + k0];
        afr[kk].y = Tm[col * 16 + k0 + 1];
    }

    // Carry powers b^(m+1) for this lane's C rows (m = r + 8*half), and b^16.
    float pcar[8];
    float p16;
    {
        const float b2 = bc * bc, b4 = b2 * b2, b8 = b4 * b4;
        p16 = b8 * b8;
        float p = half ? (b8 * bc) : bc;
#pragma unroll
        for (int r = 0; r < 8; ++r) { pcar[r] = p; p *= bc; }
    }

    const int t_out0  = q * kCHUNK;
    const int t_start = (q == 0) ? 0 : (t_out0 - kWARM);
    const int c0      = ct * 16;

    const size_t base = (size_t)bidx * kT * kC + (size_t)c0 + (size_t)col;
    // Per-lane bases with the half-wave row split baked in:
    const float* __restrict__ xph = x + base + (size_t)(2 * half) * kC;
    float* __restrict__       yph = y + base + (size_t)(8 * half) * kC;

    const int bidxp = 16 + col;     // bpermute source lane for row 15
    float yprev = 0.0f;

    // ---- Warm-up: only the carried state matters (no stores, no fix-ups).
    {
        const float* xp = xph + (size_t)t_start * kC;
        for (int t0 = t_start; t0 < t_out0; t0 += 16) {
            v2f bf[4];
            loadB(bf, xp);
            xp += (size_t)16 * kC;
            v8f c = {};
#pragma unroll
            for (int kk = 0; kk < 4; ++kk) {
                c = __builtin_amdgcn_wmma_f32_16x16x4_f32(
                    false, afr[kk], false, bf[kk], (short)0, c, false, false);
            }
            const float tail = __shfl(c[7], bidxp, 32);   // raw row 15
            yprev = fmaf(p16, yprev, tail);
        }
    }

    // ---- Main loop: software-pipelined loads, stores with immediate offsets.
    {
        const int t_end = t_out0 + kCHUNK;
        const float* xp = xph + (size_t)t_out0 * kC;
        float*       yp = yph + (size_t)t_out0 * kC;

        v2f bf[4];
        loadB(bf, xp);

        for (int t0 = t_out0; t0 < t_end; t0 += 16) {
            // Prefetch next block (clamped on the final iteration).
            const float* xn = xp + ((t0 + 16 < t_end) ? (size_t)16 * kC : 0);
            v2f bn[4];
            loadB(bn, xn);

            v8f c = {};
#pragma unroll
            for (int kk = 0; kk < 4; ++kk) {
                c = __builtin_amdgcn_wmma_f32_16x16x4_f32(
                    false, afr[kk], false, bf[kk], (short)0, c, false, false);
            }

            // Start the state shuffle early (on the raw row 15), overlap the
            // rank-1 carry fix-up with it, fold carry into the new state.
            const float tail = __shfl(c[7], bidxp, 32);
#pragma unroll
            for (int r = 0; r < 8; ++r) c[r] = fmaf(pcar[r], yprev, c[r]);
            yprev = fmaf(p16, yprev, tail);

#pragma unroll
            for (int r = 0; r < 8; ++r) {
                __builtin_nontemporal_store(c[r], yp + (size_t)r * kC);
            }

            xp = xn;
            yp += (size_t)16 * kC;
#pragma unroll
            for (int kk = 0; kk < 4; ++kk) bf[kk] = bn[kk];
        }
    }
}

extern "C" void kernel_launch(void* const* d_in, const int* in_sizes, int n_in,
                              void* d_out, int out_size, void* d_ws, size_t ws_size,
                              hipStream_t stream) {
    (void)in_sizes; (void)n_in; (void)out_size; (void)d_ws; (void)ws_size;
    const float* x = (const float*)d_in[0];
    float* yo      = (float*)d_out;
    hipLaunchKernelGGL(LowPassFilter_30580167147905_kernel,
                       dim3(kBLOCKS), dim3(256), 0, stream, x, yo);
}